// LightGraphConv_71373766525042
// MI455X (gfx1250) — compile-verified
//
#include <hip/hip_runtime.h>
#include <stdint.h>

// LightGraphConv SpMM: out[r] = sum_{e: row[e]==r} (keep[e]?val[e]:0) * x[col[e], :]
// N=100000 nodes, E=625000 edges, D=128 fp32 features.
// Roofline: 160 MFLOP vs ~650MB moved (AI ~0.24 FLOP/B) -> pure data movement,
// WMMA has no role. MI455X-rewarded paths used instead:
//  - wave32 per edge, lane owns float4 => 512B fully-coalesced gathers of x rows
//  - native fp32 global atomics for the scatter (x + out fit in the 192MB L2)
//  - edge stream (row/col/val/keep) double-buffered into LDS via
//    global_load_async_to_lds_{b32,b8} pipelined on ASYNCcnt
//  - global_prefetch_b8 lookahead on the random x-row gather

#define D_FEAT 128
#define BLOCK  256           // 8 waves of 32
#define CHUNK  256           // edges staged per block iteration (1 slot per thread)
#define CHUNKS_PER_BLOCK 4   // pipeline depth per block

__global__ void lgc_zero_kernel(float4* __restrict__ out, int n4) {
    int i = blockIdx.x * blockDim.x + threadIdx.x;
    if (i < n4) out[i] = make_float4(0.f, 0.f, 0.f, 0.f);
}

// gfx1250 async global->LDS copies (GV mode). VDST = LDS byte address,
// VADDR = 64-bit global address. Tracked by ASYNCcnt.
__device__ __forceinline__ void async_ld_b32(unsigned lds_addr, const void* gptr) {
    asm volatile("global_load_async_to_lds_b32 %0, %1, off"
                 :: "v"(lds_addr), "v"((unsigned long long)(uintptr_t)gptr)
                 : "memory");
}
__device__ __forceinline__ void async_ld_b8(unsigned lds_addr, const void* gptr) {
    asm volatile("global_load_async_to_lds_b8 %0, %1, off"
                 :: "v"(lds_addr), "v"((unsigned long long)(uintptr_t)gptr)
                 : "memory");
}

// Generic pointers to LDS carry the LDS byte offset in the low 32 bits
// (aperture occupies bits [63:32] on CDNA5).
__device__ __forceinline__ unsigned lds_lo32(const void* p) {
    return (unsigned)(uintptr_t)p;
}

__global__ __launch_bounds__(BLOCK) void lgc_spmm_scatter(
    const float* __restrict__ x,            // [N, 128]
    const float* __restrict__ vals,         // [E]
    const int*   __restrict__ rows,         // edge_index[0]
    const int*   __restrict__ cols,         // edge_index[1]
    const unsigned char* __restrict__ keep, // [E] bool
    float* __restrict__ out,                // [N, 128]
    int nEdges, int nChunksTotal)
{
    __shared__ int           s_row[2][CHUNK];
    __shared__ int           s_col[2][CHUNK];
    __shared__ float         s_val[2][CHUNK];
    __shared__ unsigned char s_keep[2][CHUNK];

    const int tid  = threadIdx.x;
    const int lane = tid & 31;
    const int wave = tid >> 5;

    const long cBase = (long)blockIdx.x * CHUNKS_PER_BLOCK;
    long cEnd = cBase + CHUNKS_PER_BLOCK;
    if (cEnd > nChunksTotal) cEnd = nChunksTotal;
    if (cBase >= cEnd) return;

    // Exactly 4 async instructions per wave per stage (tail clamped, not
    // branched) so ASYNCcnt accounting is uniform across all waves.
    auto stage = [&](int buf, long c) {
        long e  = c * CHUNK + tid;
        long ec = (e < nEdges) ? e : (long)nEdges - 1;
        async_ld_b32(lds_lo32(&s_row[buf][tid]),  rows + ec);
        async_ld_b32(lds_lo32(&s_col[buf][tid]),  cols + ec);
        async_ld_b32(lds_lo32(&s_val[buf][tid]),  vals + ec);
        async_ld_b8 (lds_lo32(&s_keep[buf][tid]), keep + ec);
    };

    // Per-edge body: broadcast LDS reads (uniform across the wave), 512B
    // coalesced gather of the x row, 4 native fp32 atomics into out.
    auto edge_body = [&](int buf, int s, bool pf) {
        const int   r  = s_row[buf][s];
        const int   cs = s_col[buf][s];
        const float v  = s_val[buf][s];

        if (pf) {  // lookahead prefetch of next edge's gathered row
            const int cn = s_col[buf][s + 1];
            __builtin_prefetch(x + (long)cn * D_FEAT + lane * 4, 0, 3);
        }

        if (s_keep[buf][s] && v != 0.0f) {   // wave-uniform branch
            const float4 xv = ((const float4*)(x + (long)cs * D_FEAT))[lane];
            float* o = out + (long)r * D_FEAT + lane * 4;
            unsafeAtomicAdd(o + 0, v * xv.x);   // global_atomic_add_f32
            unsafeAtomicAdd(o + 1, v * xv.y);
            unsafeAtomicAdd(o + 2, v * xv.z);
            unsafeAtomicAdd(o + 3, v * xv.w);
        }
    };

    stage(0, cBase);

    for (long c = cBase; c < cEnd; ++c) {
        const int buf = (int)(c - cBase) & 1;

        if (c + 1 < cEnd) {
            stage(buf ^ 1, c + 1);
            // The 4 newest (next chunk) may remain in flight; chunk c is done.
            asm volatile("s_wait_asynccnt 4" ::: "memory");
        } else {
            asm volatile("s_wait_asynccnt 0" ::: "memory");
        }
        __syncthreads();  // each wave drained its own ASYNCcnt; barrier publishes all

        const long e0  = c * CHUNK;
        const int sBeg = wave * 32;   // this wave's 32 contiguous edges in the chunk

        if (e0 + CHUNK <= nEdges) {
            // Full chunk: no per-edge bounds checks.
            #pragma unroll 4
            for (int j = 0; j < 32; ++j)
                edge_body(buf, sBeg + j, j + 1 < 32);
        } else {
            // Tail chunk only.
            for (int j = 0; j < 32; ++j) {
                if (e0 + sBeg + j >= nEdges) break;   // wave-uniform
                edge_body(buf, sBeg + j, false);
            }
        }
        __syncthreads();  // protect buffers from next iteration's async overwrite
    }
}

extern "C" void kernel_launch(void* const* d_in, const int* in_sizes, int n_in,
                              void* d_out, int out_size, void* d_ws, size_t ws_size,
                              hipStream_t stream) {
    const float*         x    = (const float*)d_in[0];
    const float*         vals = (const float*)d_in[1];
    const int*           ei   = (const int*)d_in[2];            // [2, E] row-major
    const unsigned char* keep = (const unsigned char*)d_in[3];  // bool -> 1 byte
    float*               out  = (float*)d_out;

    const int nEdges = in_sizes[1];
    const int* rows = ei;
    const int* cols = ei + nEdges;

    // Zero the output (segment_sum identity); harness poisons d_out.
    const int n4 = out_size / 4;   // 100000*128/4 float4s
    lgc_zero_kernel<<<(n4 + 255) / 256, 256, 0, stream>>>((float4*)out, n4);

    const int nChunks = (nEdges + CHUNK - 1) / CHUNK;
    const int nBlocks = (nChunks + CHUNKS_PER_BLOCK - 1) / CHUNKS_PER_BLOCK;
    lgc_spmm_scatter<<<nBlocks, BLOCK, 0, stream>>>(
        x, vals, rows, cols, keep, out, nEdges, nChunks);
}